// Model_52561809768905
// MI455X (gfx1250) — compile-verified
//
#include <hip/hip_runtime.h>
#include <hip/hip_bf16.h>
#include <math.h>

// ---- problem dims (compile-time) ----
constexpr int kLayers = 3;
constexpr int kB   = 32;
constexpr int kL   = 336;   // seq len (time)
constexpr int kN   = 321;   // nodes
constexpr int kE   = 4;     // experts
constexpr int kDFF = 512;
constexpr int kP   = 96;
constexpr int kBN  = kB * kN;          // 10272, divisible by 16
constexpr float kLossCoef = 0.01f;

typedef float v2f __attribute__((ext_vector_type(2)));
typedef float v8f __attribute__((ext_vector_type(8)));

__device__ __forceinline__ v8f wmma4(v2f a, v2f b, v8f c) {
  // D = A(16x4,f32) * B(4x16,f32) + C(16x16,f32)
  return __builtin_amdgcn_wmma_f32_16x16x4_f32(false, a, false, b, (short)0, c,
                                               false, false);
}

__device__ __forceinline__ float softplus_f(float x) {
  // jax.nn.softplus = max(x,0) + log1p(exp(-|x|))
  return fmaxf(x, 0.0f) + log1pf(expf(-fabsf(x)));
}

__device__ __forceinline__ float gelu_tanh(float h) {
  float u = 0.7978845608028654f * (h + 0.044715f * h * h * h);
  return 0.5f * h * (1.0f + tanhf(u));
}

// ------------------------------------------------------------------
// RevIN: x[B,L,N] -> X[B,N,L] normalized per (b,n) over time
// ------------------------------------------------------------------
__global__ __launch_bounds__(256) void revin_kernel(
    const float* __restrict__ x, float* __restrict__ X) {
  const int b  = blockIdx.x / 11;
  const int n0 = (blockIdx.x % 11) * 32;
  const int tx = threadIdx.x & 31;
  const int ty = threadIdx.x >> 5;     // 0..7
  const int n  = n0 + tx;
  __shared__ float ssum[8][32];
  __shared__ float ssq[8][32];
  float s = 0.f, q = 0.f;
  if (n < kN) {
    for (int l = ty; l < kL; l += 8) {
      float v = x[((size_t)b * kL + l) * kN + n];
      s += v; q += v * v;
    }
  }
  ssum[ty][tx] = s; ssq[ty][tx] = q;
  __syncthreads();
  if (ty == 0) {
    float S = 0.f, Q = 0.f;
    #pragma unroll
    for (int r = 0; r < 8; r++) { S += ssum[r][tx]; Q += ssq[r][tx]; }
    float mu  = S * (1.0f / kL);
    float var = Q * (1.0f / kL) - mu * mu;     // ddof=0
    ssum[0][tx] = mu;
    ssq[0][tx]  = 1.0f / sqrtf(var + 1e-5f);
  }
  __syncthreads();
  if (n < kN) {
    float mu = ssum[0][tx], inv = ssq[0][tx];
    for (int l = ty; l < kL; l += 8) {
      float v = x[((size_t)b * kL + l) * kN + n];
      X[((size_t)b * kN + n) * kL + l] = (v - mu) * inv;
    }
  }
}

// ------------------------------------------------------------------
// Gating: per-batch noisy top-k over E=4 experts
// ------------------------------------------------------------------
__global__ __launch_bounds__(256) void gate_kernel(
    const float* __restrict__ X, const float* __restrict__ wg,
    const float* __restrict__ wn, const float* __restrict__ noise,
    float* __restrict__ gates, float* __restrict__ loadt) {
  const int b = blockIdx.x, tid = threadIdx.x;
  __shared__ float g[kL];
  __shared__ float dots[8];
  __shared__ float cleanS[kE], nstdS[kE], noisyS[kE];

  for (int l = tid; l < kL; l += 256) {
    float s = 0.f;
    const float* xp = X + (size_t)b * kN * kL + l;
    for (int n = 0; n < kN; n++) s += xp[(size_t)n * kL];
    g[l] = s * (1.0f / kN);
  }
  __syncthreads();

  const int wave = tid >> 5, lane = tid & 31;
  {
    const int e = wave & 3;
    const float* W = (wave < 4) ? wg : wn;
    float p = 0.f;
    for (int l = lane; l < kL; l += 32) p += g[l] * W[l * kE + e];
    #pragma unroll
    for (int off = 16; off > 0; off >>= 1) p += __shfl_down(p, off, 32);
    if (lane == 0) dots[wave] = p;
  }
  __syncthreads();
  if (tid < kE) {
    float clean = dots[tid];
    float nstd  = softplus_f(dots[4 + tid]) + 1e-2f;
    cleanS[tid] = clean;
    nstdS[tid]  = nstd;
    noisyS[tid] = clean + noise[b * kE + tid] * nstd;
  }
  __syncthreads();
  if (tid == 0) {
    float v[4]; int idx[4];
    #pragma unroll
    for (int e = 0; e < 4; e++) { v[e] = noisyS[e]; idx[e] = e; }
    // stable descending sort of 4 values
    for (int i = 0; i < 3; i++)
      for (int j = 0; j < 3 - i; j++)
        if (v[j] < v[j + 1]) {
          float tv = v[j]; v[j] = v[j + 1]; v[j + 1] = tv;
          int ti = idx[j]; idx[j] = idx[j + 1]; idx[j + 1] = ti;
        }
    float e0 = expf(v[0] - v[0]), e1 = expf(v[1] - v[0]);
    float s = e0 + e1;
    float gv[4] = {0.f, 0.f, 0.f, 0.f};
    gv[idx[0]] = e0 / s; gv[idx[1]] = e1 / s;
    const float thr_in = v[2];   // (k+1)-th
    const float thr_out = v[1];  // k-th
    for (int e = 0; e < 4; e++) {
      gates[b * kE + e] = gv[e];
      bool isin = noisyS[e] > thr_in;
      float z = (cleanS[e] - (isin ? thr_in : thr_out)) / nstdS[e];
      loadt[b * kE + e] = 0.5f * (1.0f + erff(z * 0.70710678118654752f));
    }
  }
}

// ------------------------------------------------------------------
// Aux loss: cv^2(importance) + cv^2(load), accumulated across layers
// ------------------------------------------------------------------
__global__ void aux_kernel(const float* __restrict__ gates,
                           const float* __restrict__ loadt,
                           float* __restrict__ loss, int first) {
  const int tid = threadIdx.x;
  __shared__ float imp[kE], ld[kE];
  if (tid < kE) {
    float si = 0.f, sl = 0.f;
    for (int b = 0; b < kB; b++) {
      si += gates[b * kE + tid];
      sl += loadt[b * kE + tid];
    }
    imp[tid] = si; ld[tid] = sl;
  }
  __syncthreads();
  if (tid == 0) {
    auto cv2 = [](const float* v) {
      float mn = (v[0] + v[1] + v[2] + v[3]) * 0.25f;
      float var = 0.f;
      for (int e = 0; e < 4; e++) { float d = v[e] - mn; var += d * d; }
      var *= (1.0f / 3.0f);  // ddof=1
      return var / (mn * mn + 1e-10f);
    };
    float aux = kLossCoef * (cv2(imp) + cv2(ld));
    if (first) *loss = aux; else *loss += aux;
  }
}

// ------------------------------------------------------------------
// Fused MoE FFN (WMMA f32 16x16x4), in-place on X[B*N, L]
// block = 128 threads (4 waves), 16 rows per block
// ------------------------------------------------------------------
__global__ __launch_bounds__(128) void ffn_kernel(
    float* __restrict__ X, const float* __restrict__ W1,
    const float* __restrict__ b1, const float* __restrict__ W2,
    const float* __restrict__ b2, const float* __restrict__ gates) {
  __shared__ float xs[16][kL];     // 21.5 KB, input tile (residual source)
  __shared__ float hs[16][kDFF];   // 32 KB, hidden tile
  __shared__ float gs[16][kE];

  const int tid  = threadIdx.x;
  const int wave = tid >> 5;
  const int lane = tid & 31;
  const int hi   = lane >> 4;      // half-wave: selects K offset
  const int ln   = lane & 15;
  const int m0   = blockIdx.x * 16;

  for (int i = tid; i < 16 * kL; i += 128) {
    int r = i / kL, c = i % kL;
    xs[r][c] = X[(size_t)(m0 + r) * kL + c];
  }
  if (tid < 16 * kE) {
    int r = tid / kE, e = tid % kE;
    gs[r][e] = gates[((m0 + r) / kN) * kE + e];
  }
  __syncthreads();

  // persistent gated-output accumulators (init = residual)
  v8f accO[6];
  #pragma unroll
  for (int i = 0; i < 6; i++) {
    int t2 = wave + 4 * i;
    if (t2 < 21) {
      int col = t2 * 16 + ln;
      v8f a;
      #pragma unroll
      for (int j = 0; j < 8; j++) a[j] = xs[j + hi * 8][col];
      accO[i] = a;
    }
  }

  for (int e = 0; e < kE; e++) {
    const float* W1e = W1 + (size_t)e * kL * kDFF;
    const float* b1e = b1 + e * kDFF;
    const float* W2e = W2 + (size_t)e * kDFF * kL;
    const float* b2e = b2 + e * kL;
    const int c0 = wave * 128;     // this wave's 128 hidden columns

    // ---- GEMM1: H = X @ W1 (K = L) ----
    v8f accH[8];
    #pragma unroll
    for (int t = 0; t < 8; t++) accH[t] = v8f{0.f,0.f,0.f,0.f,0.f,0.f,0.f,0.f};
    for (int k = 0; k < kL; k += 4) {
      v2f a = *(const v2f*)&xs[ln][k + hi * 2];            // A frag
      const float* wrow = W1e + (size_t)(k + hi * 2) * kDFF;
      #pragma unroll
      for (int t = 0; t < 8; t++) {
        int n = c0 + t * 16 + ln;
        v2f bb; bb.x = wrow[n]; bb.y = wrow[kDFF + n];     // B frag
        accH[t] = wmma4(a, bb, accH[t]);
      }
    }
    __syncthreads();  // previous expert's hs reads are done
    #pragma unroll
    for (int t = 0; t < 8; t++) {
      int col = c0 + t * 16 + ln;
      float bias = b1e[col];
      #pragma unroll
      for (int j = 0; j < 8; j++)
        hs[j + hi * 8][col] = gelu_tanh(accH[t][j] + bias);
    }
    __syncthreads();  // hs ready

    // ---- GEMM2: Y = H @ W2 (K = DFF), gate-combine into accO ----
    #pragma unroll
    for (int i = 0; i < 6; i++) {
      int t2 = wave + 4 * i;
      if (t2 < 21) {
        int n0 = t2 * 16;
        v8f acc = v8f{0.f,0.f,0.f,0.f,0.f,0.f,0.f,0.f};
        for (int k = 0; k < kDFF; k += 4) {
          v2f a = *(const v2f*)&hs[ln][k + hi * 2];
          const float* wrow = W2e + (size_t)(k + hi * 2) * kL;
          v2f bb; bb.x = wrow[n0 + ln]; bb.y = wrow[kL + n0 + ln];
          acc = wmma4(a, bb, acc);
        }
        float bias = b2e[n0 + ln];
        #pragma unroll
        for (int j = 0; j < 8; j++)
          accO[i][j] += gs[j + hi * 8][e] * (acc[j] + bias);
      }
    }
  }
  __syncthreads();

  #pragma unroll
  for (int i = 0; i < 6; i++) {
    int t2 = wave + 4 * i;
    if (t2 < 21) {
      int col = t2 * 16 + ln;
      #pragma unroll
      for (int j = 0; j < 8; j++)
        X[(size_t)(m0 + j + hi * 8) * kL + col] = accO[i][j];
    }
  }
}

// ------------------------------------------------------------------
// Projection head (WMMA): tanh(X @ p1w^T + p1b) @ p2w^T + p2b
// block = 192 threads (6 waves), 16 rows per block
// ------------------------------------------------------------------
__global__ __launch_bounds__(192) void proj_kernel(
    const float* __restrict__ X, const float* __restrict__ p1w,
    const float* __restrict__ p1b, const float* __restrict__ p2w,
    const float* __restrict__ p2b, float* __restrict__ mean_out,
    float* __restrict__ std_out) {
  __shared__ float xs[16][kL];
  __shared__ float hsP[16][kP];

  const int tid  = threadIdx.x;
  const int wave = tid >> 5;       // 0..5
  const int lane = tid & 31;
  const int hi   = lane >> 4;
  const int ln   = lane & 15;
  const int m0   = blockIdx.x * 16;

  for (int i = tid; i < 16 * kL; i += 192) {
    int r = i / kL, c = i % kL;
    xs[r][c] = X[(size_t)(m0 + r) * kL + c];
  }
  __syncthreads();

  // GEMM1: wave w owns H columns [w*16, w*16+16)
  {
    const int p0 = wave * 16;
    v8f acc = v8f{0.f,0.f,0.f,0.f,0.f,0.f,0.f,0.f};
    for (int k = 0; k < kL; k += 4) {
      v2f a  = *(const v2f*)&xs[ln][k + hi * 2];
      v2f bb = *(const v2f*)&p1w[(size_t)(p0 + ln) * kL + k + hi * 2];
      acc = wmma4(a, bb, acc);
    }
    int col = p0 + ln;
    float bias = p1b[col];
    #pragma unroll
    for (int j = 0; j < 8; j++)
      hsP[j + hi * 8][col] = tanhf(acc[j] + bias);
  }
  __syncthreads();

  // GEMM2: 12 output tiles over 2P=192 columns; wave w -> tiles w, w+6
  #pragma unroll
  for (int ii = 0; ii < 2; ii++) {
    const int q0 = (wave + 6 * ii) * 16;
    v8f acc = v8f{0.f,0.f,0.f,0.f,0.f,0.f,0.f,0.f};
    for (int k = 0; k < kP; k += 4) {
      v2f a  = *(const v2f*)&hsP[ln][k + hi * 2];
      v2f bb = *(const v2f*)&p2w[(size_t)(q0 + ln) * kP + k + hi * 2];
      acc = wmma4(a, bb, acc);
    }
    const int q = q0 + ln;
    const float bias = p2b[q];
    const int p2 = q >> 1;
    const bool isStd = (q & 1) != 0;
    #pragma unroll
    for (int j = 0; j < 8; j++) {
      int m = m0 + j + hi * 8;
      int b = m / kN, node = m % kN;
      float val = acc[j] + bias;
      size_t oi = ((size_t)b * kP + p2) * kN + node;
      if (isStd) std_out[oi] = softplus_f(val) + 1e-6f;
      else       mean_out[oi] = val;
    }
  }
}

// ------------------------------------------------------------------
extern "C" void kernel_launch(void* const* d_in, const int* in_sizes, int n_in,
                              void* d_out, int out_size, void* d_ws,
                              size_t ws_size, hipStream_t stream) {
  (void)in_sizes; (void)n_in; (void)out_size; (void)ws_size;
  const float* x     = (const float*)d_in[0];
  const float* noise = (const float*)d_in[1];
  const float* wg    = (const float*)d_in[2];
  const float* wn    = (const float*)d_in[3];
  const float* W1    = (const float*)d_in[4];
  const float* b1    = (const float*)d_in[5];
  const float* W2    = (const float*)d_in[6];
  const float* b2    = (const float*)d_in[7];
  const float* p1w   = (const float*)d_in[8];
  const float* p1b   = (const float*)d_in[9];
  const float* p2w   = (const float*)d_in[10];
  const float* p2b   = (const float*)d_in[11];

  float* out      = (float*)d_out;
  float* X        = (float*)d_ws;                 // [B*N, L] working buffer
  float* gates    = X + (size_t)kBN * kL;         // [B, E]
  float* loadt    = gates + kB * kE;              // [B, E]
  float* mean_out = out;                          // [B, P, N]
  float* loss     = out + (size_t)kB * kP * kN;   // scalar
  float* std_out  = loss + 1;                     // [B, P, N]

  revin_kernel<<<dim3(kB * 11), dim3(256), 0, stream>>>(x, X);

  for (int l = 0; l < kLayers; l++) {
    gate_kernel<<<dim3(kB), dim3(256), 0, stream>>>(
        X, wg + (size_t)l * kL * kE, wn + (size_t)l * kL * kE,
        noise + (size_t)l * kB * kE, gates, loadt);
    aux_kernel<<<dim3(1), dim3(32), 0, stream>>>(gates, loadt, loss,
                                                 l == 0 ? 1 : 0);
    ffn_kernel<<<dim3(kBN / 16), dim3(128), 0, stream>>>(
        X, W1 + (size_t)l * kE * kL * kDFF, b1 + (size_t)l * kE * kDFF,
        W2 + (size_t)l * kE * kDFF * kL, b2 + (size_t)l * kE * kL, gates);
  }

  proj_kernel<<<dim3(kBN / 16), dim3(192), 0, stream>>>(
      X, p1w, p1b, p2w, p2b, mean_out, std_out);
}